// SimpleGNN_38173669327120
// MI455X (gfx1250) — compile-verified
//
#include <hip/hip_runtime.h>
#include <stdint.h>

#define N_NODES 100000
#define N_EDGES 1600000
#define D 128
#define LDS_PITCH 132   // 128 + 4 pad: row stride 132 dwords (528B, 16B-aligned) -> rows on distinct banks

typedef __attribute__((ext_vector_type(2))) float v2f;
typedef __attribute__((ext_vector_type(8))) float v8f;

// ---------------- degree / norm kernels ----------------

__global__ void init_deg_kernel(float* __restrict__ dout, float* __restrict__ din, int n) {
    int i = blockIdx.x * 256 + threadIdx.x;
    if (i < n) { dout[i] = 1.0f; din[i] = 1.0f; }   // self-loop contributes 1 to both
}

__global__ void count_deg_kernel(const int* __restrict__ src, const int* __restrict__ dst,
                                 float* __restrict__ dout, float* __restrict__ din, int e) {
    int i = blockIdx.x * 256 + threadIdx.x;
    if (i < e) {
        atomicAdd(&dout[src[i]], 1.0f);
        atomicAdd(&din[dst[i]], 1.0f);
    }
}

__global__ void finalize_norm_kernel(float* __restrict__ dout, float* __restrict__ din, int n) {
    int i = blockIdx.x * 256 + threadIdx.x;
    if (i < n) {                    // deg >= 1 always (self-loop), so max(.,1) is a no-op
        dout[i] = rsqrtf(dout[i]);
        din[i]  = rsqrtf(din[i]);
    }
}

// ---------------- aggregation: m = sum over edges of F[src]*norm_src[src] ----------------

// Accumulator init with the self-loop term: m[i] = F[i] * norm_src[i]
__global__ void init_m_kernel(const float* __restrict__ F, const float* __restrict__ nsrc,
                              float* __restrict__ M, int total) {
    int idx = blockIdx.x * 256 + threadIdx.x;     // over N*D
    if (idx < total) {
        int row = idx >> 7;
        M[idx] = F[idx] * nsrc[row];
    }
}

// One wave (32 lanes) per edge; each lane moves 4 consecutive floats (128 total).
__global__ __launch_bounds__(256) void aggregate_kernel(
        const float* __restrict__ F, const int* __restrict__ src, const int* __restrict__ dst,
        const float* __restrict__ nsrc, float* __restrict__ M, int e) {
    int e_id = blockIdx.x * 8 + (threadIdx.x >> 5);
    if (e_id >= e) return;
    int lane = threadIdx.x & 31;
    int s = src[e_id];
    int d = dst[e_id];
    float ns = nsrc[s];
    const float4 v = *(const float4*)(F + (size_t)s * D + lane * 4);
    float* out = M + (size_t)d * D + lane * 4;
    atomicAdd(out + 0, v.x * ns);
    atomicAdd(out + 1, v.y * ns);
    atomicAdd(out + 2, v.z * ns);
    atomicAdd(out + 3, v.w * ns);
}

// ---------------- WMMA GEMM + fused epilogue ----------------
// out[16 x 128] tile per block: 8 waves, wave w owns columns [16w, 16w+16).
// A tile (raw M rows) is async-DMA'd into padded LDS (global_load_async_to_lds_b128,
// ASYNCcnt). norm_dst is applied in the epilogue since diag(n)·(M W) == (diag(n) M) W.
// K=128 consumed in 32 chained v_wmma_f32_16x16x4_f32 steps.

template <bool BN>
__global__ __launch_bounds__(256) void gemm_epilogue_kernel(
        const float* __restrict__ M, const float* __restrict__ ndst,
        const float* __restrict__ W, const float* __restrict__ bias,
        const float* __restrict__ gamma, const float* __restrict__ beta,
        const float* __restrict__ mean, const float* __restrict__ var,
        float* __restrict__ out) {
    __shared__ float As[16 * LDS_PITCH];

    const int t = threadIdx.x;
    const int row_base = blockIdx.x * 16;

    // Async-stage the contiguous 16x128 f32 tile of M into padded LDS rows.
    // 2048 floats = 512 x b128 transfers; 256 threads x 2 issues, 16B per lane.
    #pragma unroll
    for (int i = 0; i < 2; ++i) {
        int lin = i * 1024 + t * 4;            // float index within tile
        int r = lin >> 7, c = lin & 127;
        uint32_t lds_addr = (uint32_t)(uintptr_t)&As[r * LDS_PITCH + c];
        const float* gsrc = M + (size_t)row_base * D + lin;
        asm volatile("global_load_async_to_lds_b128 %0, %1, off"
                     :: "v"(lds_addr), "v"(gsrc)
                     : "memory");
    }
    asm volatile("s_wait_asynccnt 0x0" ::: "memory");
    __syncthreads();

    const int wave = t >> 5;
    const int lane = t & 31;
    const int hi = lane >> 4;        // half-wave select (K split / row split)
    const int lo = lane & 15;        // M row for A, N col for B/C
    const int col = wave * 16 + lo;  // output column this lane owns in C layout

    const float* As_row = As + lo * LDS_PITCH;

    v8f acc = {};
    #pragma unroll
    for (int kk = 0; kk < D; kk += 4) {
        // A 16x4 f32: VGPR0 = K {kk, kk+2}, VGPR1 = K {kk+1, kk+3} across half-waves
        v2f a, b;
        a.x = As_row[kk + 2 * hi];
        a.y = As_row[kk + 2 * hi + 1];
        // B 4x16 f32: same K split, lane%16 = N; W is row-major [K][N]
        b.x = W[(size_t)(kk + 2 * hi) * D + col];
        b.y = W[(size_t)(kk + 2 * hi + 1) * D + col];
        acc = __builtin_amdgcn_wmma_f32_16x16x4_f32(
            /*neg_a=*/false, a, /*neg_b=*/false, b,
            /*c_mod=*/(short)0, acc, /*reuse_a=*/false, /*reuse_b=*/false);
    }

    // Epilogue: per-row norm_dst scale, bias (+ BN eval + leaky-relu for layer 0).
    float bb = bias[col];
    float sc = 1.0f, mu = 0.0f, bt = 0.0f;
    if (BN) {
        sc = gamma[col] * rsqrtf(var[col] + 1e-5f);
        mu = mean[col];
        bt = beta[col];
    }
    #pragma unroll
    for (int j = 0; j < 8; ++j) {
        int row = row_base + j + 8 * hi;   // C layout: VGPR j -> M = j (+8 for upper half-wave)
        float val = acc[j] * ndst[row] + bb;
        if (BN) {
            val = (val - mu) * sc + bt;
            val = val > 0.0f ? val : 0.01f * val;
        }
        out[(size_t)row * D + col] = val;
    }
}

// ---------------- launch ----------------

extern "C" void kernel_launch(void* const* d_in, const int* in_sizes, int n_in,
                              void* d_out, int out_size, void* d_ws, size_t ws_size,
                              hipStream_t stream) {
    const float* x     = (const float*)d_in[0];
    const int*   src   = (const int*)  d_in[1];
    const int*   dst   = (const int*)  d_in[2];
    const float* W1    = (const float*)d_in[3];
    const float* b1    = (const float*)d_in[4];
    const float* W2    = (const float*)d_in[5];
    const float* b2    = (const float*)d_in[6];
    const float* gamma = (const float*)d_in[7];
    const float* beta  = (const float*)d_in[8];
    const float* rmean = (const float*)d_in[9];
    const float* rvar  = (const float*)d_in[10];
    float* out = (float*)d_out;

    float* bufA = (float*)d_ws;                       // [N, 128] layer-0 output
    float* bufB = bufA + (size_t)N_NODES * D;         // [N, 128] aggregation accumulator
    float* nsrc = bufB + (size_t)N_NODES * D;         // [N] out-deg^-1/2
    float* ndst = nsrc + N_NODES;                     // [N] in-deg^-1/2

    dim3 blk(256);

    // degrees -> norms
    init_deg_kernel     <<<(N_NODES + 255) / 256, blk, 0, stream>>>(nsrc, ndst, N_NODES);
    count_deg_kernel    <<<(N_EDGES + 255) / 256, blk, 0, stream>>>(src, dst, nsrc, ndst, N_EDGES);
    finalize_norm_kernel<<<(N_NODES + 255) / 256, blk, 0, stream>>>(nsrc, ndst, N_NODES);

    const int total = N_NODES * D;

    // layer 0: aggregate(x*nsrc) -> diag(ndst)·(M@W1) + b1 -> BN -> leaky  => bufA
    init_m_kernel<<<(total + 255) / 256, blk, 0, stream>>>(x, nsrc, bufB, total);
    aggregate_kernel<<<(N_EDGES + 7) / 8, blk, 0, stream>>>(x, src, dst, nsrc, bufB, N_EDGES);
    gemm_epilogue_kernel<true><<<N_NODES / 16, blk, 0, stream>>>(
        bufB, ndst, W1, b1, gamma, beta, rmean, rvar, bufA);

    // layer 1: aggregate(bufA*nsrc) -> diag(ndst)·(M@W2) + b2  => d_out
    init_m_kernel<<<(total + 255) / 256, blk, 0, stream>>>(bufA, nsrc, bufB, total);
    aggregate_kernel<<<(N_EDGES + 7) / 8, blk, 0, stream>>>(bufA, src, dst, nsrc, bufB, N_EDGES);
    gemm_epilogue_kernel<false><<<N_NODES / 16, blk, 0, stream>>>(
        bufB, ndst, W2, b2, nullptr, nullptr, nullptr, nullptr, out);
}